// Customlosskll1_90829968376293
// MI455X (gfx1250) — compile-verified
//
#include <hip/hip_runtime.h>

// CDNA5 / gfx1250, wave32. One workgroup per histogram row; LDS fixed-point
// histograms (deterministic); wave reductions via V_WMMA_F32_16X16X4_F32 +
// ds_swizzle lane folds.

typedef __attribute__((ext_vector_type(2))) float v2f;
typedef __attribute__((ext_vector_type(8))) float v8f;

#define N_BINS 100
#define HSCALE 2097152.0f          // 2^21 fixed-point scale for histogram weights
#define INV_HSCALE (1.0f / 2097152.0f)

// Sum of all 32 lane values, returned in every lane.
// D[m][n] = sum_k A[m][k]*B[k][n]; with A == all-ones (covers every A slot however
// lanes map), D[m][n] = column-n sum of B. B's two VGPRs across 32 lanes cover all
// 64 B slots bijectively; values go in b[0], zeros in b[1], so the 16 column sums
// of D partition the 32 lane values. Documented C/D layout: d[0] at lane L holds
// D[row][L%16], row-independent here -> 4 xor-swizzles finish the sum.
__device__ __forceinline__ float wave_sum32(float v) {
    v2f a; a[0] = 1.0f; a[1] = 1.0f;      // A = ones(16x4)
    v2f b; b[0] = v;    b[1] = 0.0f;      // 32 values + 32 zeros fill B(4x16)
    v8f c = {};
    c = __builtin_amdgcn_wmma_f32_16x16x4_f32(false, a, false, b, (short)0, c, false, false);
    float t = c[0];                       // lane L: colsum[L % 16]
    // ds_swizzle group-of-32: offset = (xor<<10) | (or<<5) | and ; and=0x1f, or=0
    t += __int_as_float(__builtin_amdgcn_ds_swizzle(__float_as_int(t), 0x201F)); // xor 8
    t += __int_as_float(__builtin_amdgcn_ds_swizzle(__float_as_int(t), 0x101F)); // xor 4
    t += __int_as_float(__builtin_amdgcn_ds_swizzle(__float_as_int(t), 0x081F)); // xor 2
    t += __int_as_float(__builtin_amdgcn_ds_swizzle(__float_as_int(t), 0x041F)); // xor 1
    return t;
}

// Block-wide sum, broadcast to all threads. red[] must hold >= NW floats.
// Cross-wave combine is a fixed-order unrolled sum -> deterministic.
template <int NW>
__device__ __forceinline__ float block_sum_bcast(float v, float* red) {
    const float w  = wave_sum32(v);
    const int lane = (int)threadIdx.x & 31;
    const int wid  = (int)threadIdx.x >> 5;
    __syncthreads();
    if (lane == 0) red[wid] = w;
    __syncthreads();
    float t = 0.0f;
#pragma unroll
    for (int i = 0; i < NW; ++i) t += red[i];
    return t;
}

// Linear-interpolation (triangular kernel) histogram splat, fixed-point.
// s = x/BW - 0.5; bin floor(s) gets (1-f), bin floor(s)+1 gets f.
__device__ __forceinline__ void bin_add(unsigned int* hist, float x) {
    const float s  = x * 100.0f - 0.5f;
    const float fk = floorf(s);
    const int   k0 = (int)fk;
    const float f  = s - fk;
    const unsigned int wHi = (unsigned int)(f * HSCALE + 0.5f);
    const unsigned int wLo = (unsigned int)((1.0f - f) * HSCALE + 0.5f);
    if ((unsigned)k0       < (unsigned)N_BINS) atomicAdd(&hist[k0],     wLo);
    if ((unsigned)(k0 + 1) < (unsigned)N_BINS) atomicAdd(&hist[k0 + 1], wHi);
}

// One block per (b,h) row: builds 100-bin pdfs of x-row and t-row, KL(pc||pn),
// and the row's sum of |x-t| for parta. 256 thr = 8 waves, float2 loads.
__global__ __launch_bounds__(256) void row_kernel(const float* __restrict__ x,
                                                  const float* __restrict__ t,
                                                  float* __restrict__ out_diff,
                                                  float* __restrict__ out_kl) {
    __shared__ unsigned int hx[N_BINS];
    __shared__ unsigned int ht[N_BINS];
    __shared__ float red[8];
    const int  tid = (int)threadIdx.x;
    const long row = (long)blockIdx.x;                 // b*512 + h
    const float2* __restrict__ xr = (const float2*)(x + row * 512);
    const float2* __restrict__ tr = (const float2*)(t + row * 512);

    if (tid < N_BINS) { hx[tid] = 0u; ht[tid] = 0u; }

    const float2 xv = xr[tid];
    const float2 tv = tr[tid];
    __syncthreads();

    const float dsum = fabsf(xv.x - tv.x) + fabsf(xv.y - tv.y);
    bin_add(hx, xv.x);
    bin_add(hx, xv.y);
    bin_add(ht, tv.x);
    bin_add(ht, tv.y);
    __syncthreads();

    const float hxv = (tid < N_BINS) ? (float)hx[tid] * INV_HSCALE : 0.0f;
    const float htv = (tid < N_BINS) ? (float)ht[tid] * INV_HSCALE : 0.0f;
    const float sumX = block_sum_bcast<8>(hxv, red);
    const float sumT = block_sum_bcast<8>(htv, red);

    float term = 0.0f;
    if (tid < N_BINS) {
        const float pn = hxv / (sumX + 1e-10f) + 1e-5f;
        const float pc = htv / (sumT + 1e-10f) + 1e-5f;
        term = pc * (logf(pc) - logf(pn));
    }
    const float kl   = block_sum_bcast<8>(term, red);
    const float dtot = block_sum_bcast<8>(dsum, red);
    if (tid == 0) { out_diff[row] = dtot; out_kl[row] = kl; }
}

// One block per batch: histogram/KL over the first column x[b,:,0] (stride 512).
__global__ __launch_bounds__(512) void col_kernel(const float* __restrict__ x,
                                                  const float* __restrict__ t,
                                                  float* __restrict__ out_kl) {
    __shared__ unsigned int hx[N_BINS];
    __shared__ unsigned int ht[N_BINS];
    __shared__ float red[16];
    const int tid = (int)threadIdx.x;                  // h = 0..511
    const int b   = (int)blockIdx.x;
    const size_t base = (size_t)b * 262144 + (size_t)tid * 512;
    const float xv = x[base];
    const float tv = t[base];

    if (tid < N_BINS) { hx[tid] = 0u; ht[tid] = 0u; }
    __syncthreads();
    bin_add(hx, xv);
    bin_add(ht, tv);
    __syncthreads();

    const float hxv = (tid < N_BINS) ? (float)hx[tid] * INV_HSCALE : 0.0f;
    const float htv = (tid < N_BINS) ? (float)ht[tid] * INV_HSCALE : 0.0f;
    const float sumX = block_sum_bcast<16>(hxv, red);
    const float sumT = block_sum_bcast<16>(htv, red);

    float term = 0.0f;
    if (tid < N_BINS) {
        const float pn = hxv / (sumX + 1e-10f) + 1e-6f;   // +eps (1e-6) for columns
        const float pc = htv / (sumT + 1e-10f) + 1e-6f;
        term = pc * (logf(pc) - logf(pn));
    }
    const float kl = block_sum_bcast<16>(term, red);
    if (tid == 0) out_kl[b] = kl;
}

// ws layout (floats): [0..2047] row diff sums | [2048..4095] row KLs | [4096..4099] col KLs
__global__ __launch_bounds__(256) void final_kernel(const float* __restrict__ ws,
                                                    const float* __restrict__ we1,
                                                    const float* __restrict__ we2,
                                                    const float* __restrict__ we3,
                                                    float* __restrict__ out) {
    __shared__ float red[8];
    const int tid = (int)threadIdx.x;
    float sA = 0.0f, sB = 0.0f;
#pragma unroll
    for (int j = 0; j < 8; ++j) {
        const int   i  = tid + j * 256;
        const int   b  = i >> 9;                     // i / 512
        const float w1 = we1[b] + 1e-6f;
        const float w2 = we2[b] + 1e-6f;
        sA += ws[i]        * (w1 + 1.0f / w1);
        sB += ws[2048 + i] * (w2 + 1.0f / w2);
    }
    const float tA = block_sum_bcast<8>(sA, red);
    const float tB = block_sum_bcast<8>(sB, red);
    if (tid == 0) {
        float sC = 0.0f;
        for (int b = 0; b < 4; ++b) {
            const float w3 = we3[b] + 1e-6f;
            sC += ws[4096 + b] * (w3 + 1.0f / w3);
        }
        const float parta = tA * (1.0f / 1048576.0f);              // mean over 4*512*512
        const float partb = (tB * (1.0f / 2048.0f) + sC * (1.0f / 2048.0f)) * 0.5f;
        out[0] = parta + partb;
    }
}

extern "C" void kernel_launch(void* const* d_in, const int* in_sizes, int n_in,
                              void* d_out, int out_size, void* d_ws, size_t ws_size,
                              hipStream_t stream) {
    (void)in_sizes; (void)n_in; (void)out_size; (void)ws_size;
    const float* inputo = (const float*)d_in[0];
    const float* target = (const float*)d_in[1];
    const float* we1    = (const float*)d_in[2];
    const float* we2    = (const float*)d_in[3];
    const float* we3    = (const float*)d_in[4];
    float* out = (float*)d_out;
    float* ws  = (float*)d_ws;

    row_kernel<<<2048, 256, 0, stream>>>(inputo, target, ws, ws + 2048);
    col_kernel<<<4, 512, 0, stream>>>(inputo, target, ws + 4096);
    final_kernel<<<1, 256, 0, stream>>>(ws, we1, we2, we3, out);
}